// Net_52037823758875
// MI455X (gfx1250) — compile-verified
//
#include <hip/hip_runtime.h>
#include <hip/hip_bf16.h>

// GCN 2-layer inference for MI455X (gfx1250, wave32, WMMA + TDM).
// Roofline: edge scatter moves ~4 GB of L2/HBM atomic traffic (~200us @ 23.3 TB/s);
// GEMMs (6.5 + 0.5 GFLOP) are microseconds on the matrix pipe -> use full-precision
// V_WMMA_F32_16X16X4_F32 (cheap precision buys nothing here).
// W-panel staging into LDS uses the Tensor Data Mover (tensor_load_to_lds) when the
// builtin is available; scatter uses hardware global_atomic_add_f32 (unsafeAtomicAdd).

typedef __attribute__((ext_vector_type(2))) float v2f;
typedef __attribute__((ext_vector_type(8))) float v8f;
typedef __attribute__((ext_vector_type(4))) unsigned int u32x4;
typedef __attribute__((ext_vector_type(8))) int i32x8;
typedef __attribute__((ext_vector_type(4))) int i32x4;

constexpr int NN  = 100000;   // nodes
constexpr int IND = 512;      // input features
constexpr int HID = 64;       // hidden features
constexpr int NC  = 40;       // classes
constexpr int NCP = 48;       // classes padded to 3 WMMA n-tiles

#if defined(__has_builtin)
#if __has_builtin(__builtin_amdgcn_tensor_load_to_lds)
#define HAVE_TDM 1
#endif
#endif

#ifdef HAVE_TDM
// Issue one TDM 2D tile load: nrows x rowlen_elts f32, row stride stride_elts,
// from gaddr into LDS byte offset lds_off. Descriptor per CDNA5 ISA ch.8.
__device__ __forceinline__ void tdm_load_2d_f32(unsigned lds_off, const void* gaddr,
                                                unsigned rowlen_elts, unsigned nrows,
                                                unsigned stride_elts, unsigned tensor_rows) {
  unsigned long long ga = (unsigned long long)gaddr;
  u32x4 g0;
  g0[0] = 1u;                                    // count=1 valid user descriptor
  g0[1] = lds_off;                               // lds_addr (bytes)
  g0[2] = (unsigned)(ga & 0xffffffffu);          // global_addr[31:0]
  g0[3] = (unsigned)((ga >> 32) & 0x01ffffffu)   // global_addr[56:32]
        | (2u << 30);                            // type=2 ("image")
  i32x8 g1;
  g1[0] = (int)(2u << 16);                       // data_size=4B; mask/pad/iter=0
  g1[1] = (int)((rowlen_elts & 0xffffu) << 16);  // tensor_dim0[15:0]
  g1[2] = (int)((tensor_rows & 0xffffu) << 16);  // tensor_dim0[31:16]=0 | tensor_dim1[15:0]
  g1[3] = (int)((rowlen_elts & 0xffffu) << 16);  // tensor_dim1[31:16]=0 | tile_dim0
  g1[4] = (int)(nrows & 0xffffu);                // tile_dim1 | tile_dim2=0 (2D)
  g1[5] = (int)stride_elts;                      // tensor_dim0_stride[31:0]
  g1[6] = 0;                                     // stride0[47:32] | stride1[15:0]
  g1[7] = 0;                                     // stride1[47:16]
  i32x4 z4 = {0, 0, 0, 0};
#if defined(__clang_major__) && (__clang_major__ >= 23)
  i32x8 z8 = {0, 0, 0, 0, 0, 0, 0, 0};
  __builtin_amdgcn_tensor_load_to_lds(g0, g1, z4, z4, z8, 0);
#else
  __builtin_amdgcn_tensor_load_to_lds(g0, g1, z4, z4, 0);
#endif
#if __has_builtin(__builtin_amdgcn_s_wait_tensorcnt)
  __builtin_amdgcn_s_wait_tensorcnt(0);
#else
  asm volatile("s_wait_tensorcnt 0x0" ::: "memory");
#endif
}
#endif  // HAVE_TDM

// ---------------- degree / symmetric norm ----------------

__global__ void k_init_deg(float* __restrict__ deg) {
  int i = blockIdx.x * blockDim.x + threadIdx.x;
  if (i < NN) deg[i] = 1.0f;  // self-loop contribution
}

__global__ void k_count_deg(const long long* __restrict__ dst,
                            float* __restrict__ deg, int ne) {
  int i = blockIdx.x * blockDim.x + threadIdx.x;
  if (i < ne) unsafeAtomicAdd(&deg[(int)dst[i]], 1.0f);
}

__global__ void k_dinv(float* __restrict__ deg) {
  int i = blockIdx.x * blockDim.x + threadIdx.x;
  if (i < NN) deg[i] = rsqrtf(deg[i]);  // deg >= 1 always (self-loop)
}

// ---------------- GEMM1: H1 = X (100000x512) @ W1 (512x64) ----------------
// One wave per 16-row M-tile; W1 staged in LDS as 128x64 K-panels (32 KB) via TDM.
// A fragment (16x4 f32): lanes 0-15 carry K=k,k+1; lanes 16-31 carry K=k+2,k+3.
// C/D (16x16 f32): VGPR j = row j (lanes 0-15) / row j+8 (lanes 16-31), col = lane&15.

__global__ __launch_bounds__(64) void k_gemm1(const float* __restrict__ X,
                                              const float* __restrict__ W,
                                              float* __restrict__ H) {
  __shared__ float Ws[128 * HID];  // 32 KB K-panel of W1
  const int lane  = threadIdx.x & 31;
  const int wave  = threadIdx.x >> 5;
  const int mtile = blockIdx.x * 2 + wave;       // 3125 blocks * 2 waves = 6250 tiles
  const size_t row0 = (size_t)mtile * 16;
  const int lhalf = (lane < 16) ? 0 : 2;
  const int l15   = lane & 15;

  v8f acc[4] = {};

  for (int kp = 0; kp < IND; kp += 128) {
    __syncthreads();
#ifdef HAVE_TDM
    if (wave == 0) {
      tdm_load_2d_f32((unsigned)(uintptr_t)Ws, W + (size_t)kp * HID,
                      HID, 128, HID, IND);
    }
#else
    {
      const float4* gsrc = (const float4*)(W + (size_t)kp * HID);
      for (int i = threadIdx.x; i < 128 * HID / 4; i += 64)
        ((float4*)Ws)[i] = gsrc[i];
    }
#endif
    __syncthreads();

    const float* xrow = X + (row0 + l15) * IND + kp;
#pragma unroll 4
    for (int k = 0; k < 128; k += 4) {
      v2f a;
      a.x = xrow[k + lhalf];
      a.y = xrow[k + lhalf + 1];
#pragma unroll
      for (int nt = 0; nt < 4; ++nt) {
        v2f b;
        b.x = Ws[(k + lhalf)     * HID + nt * 16 + l15];
        b.y = Ws[(k + lhalf + 1) * HID + nt * 16 + l15];
        acc[nt] = __builtin_amdgcn_wmma_f32_16x16x4_f32(
            false, a, false, b, (short)0, acc[nt], false, false);
      }
    }
  }

  const size_t rbase = row0 + ((lane < 16) ? 0 : 8);
#pragma unroll
  for (int nt = 0; nt < 4; ++nt)
#pragma unroll
    for (int j = 0; j < 8; ++j)
      H[(rbase + j) * HID + nt * 16 + l15] = acc[nt][j];
}

// ---------------- GEMM2: H2 = AGG1 (100000x64) @ W2 (64x40) ----------------

__global__ __launch_bounds__(64) void k_gemm2(const float* __restrict__ A,
                                              const float* __restrict__ W,
                                              float* __restrict__ H) {
  __shared__ float Ws[HID * NCP];  // 64x48, zero-padded cols 40..47 (12 KB)
  for (int i = threadIdx.x; i < HID * NCP; i += 64) {
    int r = i / NCP, c = i % NCP;
    Ws[i] = (c < NC) ? W[r * NC + c] : 0.0f;
  }
  __syncthreads();

  const int lane  = threadIdx.x & 31;
  const int wave  = threadIdx.x >> 5;
  const int mtile = blockIdx.x * 2 + wave;
  const size_t row0 = (size_t)mtile * 16;
  const int lhalf = (lane < 16) ? 0 : 2;
  const int l15   = lane & 15;

  v8f acc[3] = {};
  const float* arow = A + (row0 + l15) * HID;
#pragma unroll
  for (int k = 0; k < HID; k += 4) {
    v2f a;
    a.x = arow[k + lhalf];
    a.y = arow[k + lhalf + 1];
#pragma unroll
    for (int nt = 0; nt < 3; ++nt) {
      v2f b;
      b.x = Ws[(k + lhalf)     * NCP + nt * 16 + l15];
      b.y = Ws[(k + lhalf + 1) * NCP + nt * 16 + l15];
      acc[nt] = __builtin_amdgcn_wmma_f32_16x16x4_f32(
          false, a, false, b, (short)0, acc[nt], false, false);
    }
  }

  const size_t rbase = row0 + ((lane < 16) ? 0 : 8);
#pragma unroll
  for (int nt = 0; nt < 3; ++nt) {
    int col = nt * 16 + l15;
    if (col < NC)
#pragma unroll
      for (int j = 0; j < 8; ++j)
        H[(rbase + j) * NC + col] = acc[nt][j];
  }
}

// ---------------- message passing (gather * norm, scatter-add) ----------------

__global__ void k_self1(const float* __restrict__ H, const float* __restrict__ dinv,
                        float* __restrict__ AGG) {
  int i = blockIdx.x * blockDim.x + threadIdx.x;  // NN*16 float4 chunks
  if (i >= NN * 16) return;
  int n = i >> 4, f = (i & 15) << 2;
  float w = dinv[n] * dinv[n];
  float4 h = *(const float4*)(H + (size_t)n * HID + f);
  float4 o = { h.x * w, h.y * w, h.z * w, h.w * w };
  *(float4*)(AGG + (size_t)n * HID + f) = o;   // initializes AGG (no pre-zero needed)
}

__global__ void k_scatter1(const long long* __restrict__ src, const long long* __restrict__ dst,
                           const float* __restrict__ dinv, const float* __restrict__ H,
                           float* __restrict__ AGG, int ne) {
  long long tid = (long long)blockIdx.x * blockDim.x + threadIdx.x;
  if (tid >= (long long)ne * 16) return;
  int e = (int)(tid >> 4), f = ((int)tid & 15) << 2;
  int s = (int)src[e], d = (int)dst[e];
  float w = dinv[s] * dinv[d];
  float4 h = *(const float4*)(H + (size_t)s * HID + f);
  float* o = AGG + (size_t)d * HID + f;
  unsafeAtomicAdd(o + 0, h.x * w);
  unsafeAtomicAdd(o + 1, h.y * w);
  unsafeAtomicAdd(o + 2, h.z * w);
  unsafeAtomicAdd(o + 3, h.w * w);
}

__global__ void k_bias_relu(float* __restrict__ A, const float* __restrict__ b) {
  int i = blockIdx.x * blockDim.x + threadIdx.x;
  if (i < NN * HID) {
    float v = A[i] + b[i & (HID - 1)];
    A[i] = v > 0.0f ? v : 0.0f;
  }
}

__global__ void k_self2(const float* __restrict__ H, const float* __restrict__ dinv,
                        float* __restrict__ AGG) {
  int i = blockIdx.x * blockDim.x + threadIdx.x;  // NN*10 float4 chunks (40 floats/row)
  if (i >= NN * 10) return;
  int n = i / 10, f = (i % 10) << 2;
  float w = dinv[n] * dinv[n];
  float4 h = *(const float4*)(H + (size_t)n * NC + f);
  float4 o = { h.x * w, h.y * w, h.z * w, h.w * w };
  *(float4*)(AGG + (size_t)n * NC + f) = o;
}

__global__ void k_scatter2(const long long* __restrict__ src, const long long* __restrict__ dst,
                           const float* __restrict__ dinv, const float* __restrict__ H,
                           float* __restrict__ AGG, int ne) {
  long long tid = (long long)blockIdx.x * blockDim.x + threadIdx.x;
  if (tid >= (long long)ne * 10) return;
  int e = (int)(tid / 10), f = ((int)(tid % 10)) << 2;
  int s = (int)src[e], d = (int)dst[e];
  float w = dinv[s] * dinv[d];
  float4 h = *(const float4*)(H + (size_t)s * NC + f);
  float* o = AGG + (size_t)d * NC + f;
  unsafeAtomicAdd(o + 0, h.x * w);
  unsafeAtomicAdd(o + 1, h.y * w);
  unsafeAtomicAdd(o + 2, h.z * w);
  unsafeAtomicAdd(o + 3, h.w * w);
}

// ---------------- bias + log_softmax ----------------

__global__ void k_logsoftmax(const float* __restrict__ A, const float* __restrict__ b,
                             float* __restrict__ out) {
  int n = blockIdx.x * blockDim.x + threadIdx.x;
  if (n >= NN) return;
  float v[NC];
  float m = -1e30f;
#pragma unroll
  for (int c = 0; c < NC; ++c) {
    v[c] = A[(size_t)n * NC + c] + b[c];
    m = fmaxf(m, v[c]);
  }
  float s = 0.0f;
#pragma unroll
  for (int c = 0; c < NC; ++c) s += __expf(v[c] - m);
  float l = __logf(s) + m;
#pragma unroll
  for (int c = 0; c < NC; ++c) out[(size_t)n * NC + c] = v[c] - l;
}

// ---------------- launch ----------------

extern "C" void kernel_launch(void* const* d_in, const int* in_sizes, int n_in,
                              void* d_out, int out_size, void* d_ws, size_t ws_size,
                              hipStream_t stream) {
  const float*     X  = (const float*)d_in[0];
  const long long* ei = (const long long*)d_in[1];   // int64 edge_index [2][E]
  const float*     W1 = (const float*)d_in[2];
  const float*     b1 = (const float*)d_in[3];
  const float*     W2 = (const float*)d_in[4];
  const float*     b2 = (const float*)d_in[5];
  float*           out = (float*)d_out;

  const int ne = in_sizes[1] / 2;
  const long long* src = ei;
  const long long* dst = ei + ne;

  // scratch layout (~84 MB total)
  char* ws = (char*)d_ws;
  auto alloc = [&](size_t bytes) {
    char* p = ws;
    ws += (bytes + 255) & ~(size_t)255;
    return (float*)p;
  };
  float* dinv = alloc(sizeof(float) * NN);            // deg -> dinv in place
  float* H1   = alloc(sizeof(float) * NN * HID);
  float* AG1  = alloc(sizeof(float) * NN * HID);
  float* H2   = alloc(sizeof(float) * NN * NC);
  float* AG2  = alloc(sizeof(float) * NN * NC);

  k_init_deg <<<(NN + 255) / 256, 256, 0, stream>>>(dinv);
  k_count_deg<<<(ne + 255) / 256, 256, 0, stream>>>(dst, dinv, ne);
  k_dinv     <<<(NN + 255) / 256, 256, 0, stream>>>(dinv);

  k_gemm1<<<NN / 32, 64, 0, stream>>>(X, W1, H1);     // 3125 blocks x 2 waves

  k_self1<<<(NN * 16 + 255) / 256, 256, 0, stream>>>(H1, dinv, AG1);
  {
    long long t = (long long)ne * 16;
    k_scatter1<<<(int)((t + 255) / 256), 256, 0, stream>>>(src, dst, dinv, H1, AG1, ne);
  }
  k_bias_relu<<<(NN * HID + 255) / 256, 256, 0, stream>>>(AG1, b1);

  k_gemm2<<<NN / 32, 64, 0, stream>>>(AG1, W2, H2);

  k_self2<<<(NN * 10 + 255) / 256, 256, 0, stream>>>(H2, dinv, AG2);
  {
    long long t = (long long)ne * 10;
    k_scatter2<<<(int)((t + 255) / 256), 256, 0, stream>>>(src, dst, dinv, H2, AG2, ne);
  }
  k_logsoftmax<<<(NN + 255) / 256, 256, 0, stream>>>(AG2, b2, out);
}